// xCPE_687194768153
// MI455X (gfx1250) — compile-verified
//
#include <hip/hip_runtime.h>
#include <hip/hip_bf16.h>
#include <math.h>

// ---------------------------------------------------------------------------
// Fused point-cloud CPE block for MI455X (gfx1250, wave32).
//   pass 1: per-scene xyz mean (8 blocks, tiny)
//   pass 2: fused  Linear(3,256)+LN+GELU -> +feat -> depthwise conv k=3 -> +feat -> LN
//           Linear done with V_WMMA_F32_16X16X4_F32 (bias folded into K=3).
// Memory-bound: one feat read + one out write per point (~0.52 GB total).
// v2: per-row precompute stage kills the divergent seg_of loops in the hot
//     phases; lane->channel mapping changed to (lane*4, 128+lane*4) to halve
//     LDS bank conflicts on the b128 row accesses.
// ---------------------------------------------------------------------------

#define C_DIM   256
#define TPB     256          // 8 waves of 32
#define F_PTS   48           // f rows computed per block (3 WMMA point-tiles)
#define OUT_PTS 46           // interior outputs per block (halo of 1 each side)
#define MAXB    16

typedef __attribute__((ext_vector_type(2))) float v2f;
typedef __attribute__((ext_vector_type(8))) float v8f;

__device__ __forceinline__ float wave_sum32(float v) {
#pragma unroll
    for (int m = 16; m >= 1; m >>= 1) v += __shfl_xor(v, m, 32);
    return v;
}

__device__ __forceinline__ int seg_of(long long g, const long long* soff, int Bn) {
    int s = 0;
    while (s + 1 < Bn && g >= soff[s + 1]) ++s;
    return s;
}

// ---------------- pass 1: per-scene coordinate means -----------------------
__global__ void scene_mean_kernel(const float* __restrict__ coord,
                                  const long long* __restrict__ lengths,
                                  float* __restrict__ means, int Bn) {
    const int s = blockIdx.x;
    long long off = 0;
    for (int i = 0; i < s; ++i) off += lengths[i];
    const long long len = lengths[s];

    float sx = 0.f, sy = 0.f, sz = 0.f;
    for (long long i = threadIdx.x; i < len; i += blockDim.x) {
        const long long g = off + i;
        sx += coord[3 * g + 0];
        sy += coord[3 * g + 1];
        sz += coord[3 * g + 2];
    }
    __shared__ float rb[3][TPB];
    rb[0][threadIdx.x] = sx; rb[1][threadIdx.x] = sy; rb[2][threadIdx.x] = sz;
    __syncthreads();
    for (int st = TPB / 2; st > 0; st >>= 1) {
        if (threadIdx.x < st) {
            rb[0][threadIdx.x] += rb[0][threadIdx.x + st];
            rb[1][threadIdx.x] += rb[1][threadIdx.x + st];
            rb[2][threadIdx.x] += rb[2][threadIdx.x + st];
        }
        __syncthreads();
    }
    if (threadIdx.x == 0) {
        const float inv = 1.f / (float)(len > 0 ? len : 1);
        means[3 * s + 0] = rb[0][0] * inv;
        means[3 * s + 1] = rb[1][0] * inv;
        means[3 * s + 2] = rb[2][0] * inv;
    }
}

// ---------------- pass 2: fused main kernel --------------------------------
__global__ void __launch_bounds__(TPB)
cpe_fused_kernel(const float* __restrict__ feat,
                 const float* __restrict__ coord,
                 const long long* __restrict__ lengths,
                 const float* __restrict__ w_xyz,   // [3, C]
                 const float* __restrict__ b_xyz,   // [C]
                 const float* __restrict__ g1,
                 const float* __restrict__ be1,
                 const float* __restrict__ w_conv,  // [C, 3]
                 const float* __restrict__ b_conv,
                 const float* __restrict__ g2,
                 const float* __restrict__ be2,
                 const float* __restrict__ means,   // [B, 3]
                 float* __restrict__ out,
                 int N, int Bn) {
    __shared__ float     lin[F_PTS][C_DIM];       // 48 KiB staging: lin -> f
    __shared__ long long soff[MAXB + 1];
    __shared__ float     smean[MAXB][3];
    __shared__ float4    sxyzw[F_PTS];            // centered xyz + valid flag
    __shared__ float     spm[F_PTS], snm[F_PTS];  // conv prev/next masks

    const int tid  = threadIdx.x;
    const int wv   = tid >> 5;        // wave id 0..7
    const int lane = tid & 31;
    const int lh   = lane & 15;
    const bool hi  = lane >= 16;
    const int base = blockIdx.x * OUT_PTS - 1;    // global index of f row 0

    // ---- stage A: segment offsets + scene means into LDS
    if (tid == 0) {
        soff[0] = 0;
        for (int i = 0; i < Bn && i < MAXB; ++i) soff[i + 1] = soff[i] + lengths[i];
    }
    if (tid < Bn * 3 && tid < MAXB * 3) smean[tid / 3][tid % 3] = means[tid];
    __syncthreads();

    // ---- stage B: per-row precompute (one thread per f row; seg walk once)
    if (tid < F_PTS) {
        const int g = base + tid;
        const bool valid = (g >= 0) && (g < N);
        const int gc = valid ? g : 0;
        const int s  = seg_of(gc, soff, Bn);
        float4 xw;
        xw.x = valid ? (coord[3 * gc + 0] - smean[s][0]) : 0.f;
        xw.y = valid ? (coord[3 * gc + 1] - smean[s][1]) : 0.f;
        xw.z = valid ? (coord[3 * gc + 2] - smean[s][2]) : 0.f;
        xw.w = valid ? 1.f : 0.f;                  // multiplies the bias row (K=3)
        sxyzw[tid] = xw;
        spm[tid] = (valid && g > (long long)soff[s])     ? 1.f : 0.f;
        snm[tid] = (valid && (long long)g + 1 < soff[s + 1]) ? 1.f : 0.f;
    }
    __syncthreads();

    // ---- phase 1: pos_lin[48][256] = [xyz_c,1] @ [W;b]  via WMMA f32 16x16x4
    // A (16x4, MxK): lanes 0-15 hold K=0,1 ; lanes 16-31 hold K=2,3 ; M = lane&15
    // B (4x16, KxN): lanes 0-15 hold K=0,1 ; lanes 16-31 hold K=2,3 ; N = lane&15
    // D (16x16):     lanes 0-15 -> M = v ; lanes 16-31 -> M = v+8 ; N = lane&15
    for (int pt = 0; pt < 3; ++pt) {
        const float4 xw = sxyzw[pt * 16 + lh];
        v2f a;
        a.x = hi ? xw.z : xw.x;
        a.y = hi ? xw.w : xw.y;

        for (int ct = 2 * wv; ct < 2 * wv + 2; ++ct) {
            const int n = ct * 16 + lh;
            v2f bm;
            bm.x = hi ? w_xyz[2 * C_DIM + n] : w_xyz[0 * C_DIM + n];
            bm.y = hi ? b_xyz[n]             : w_xyz[1 * C_DIM + n];
            v8f c = {};
            v8f d = __builtin_amdgcn_wmma_f32_16x16x4_f32(
                false, a, false, bm, (short)0, c, false, false);
            const int rowb = pt * 16 + (hi ? 8 : 0);
#pragma unroll
            for (int v = 0; v < 8; ++v) lin[rowb + v][n] = d[v];
        }
    }
    __syncthreads();

    // lane -> channel mapping: chunk0 = lane*4 .. +3, chunk1 = 128 + lane*4 .. +3
    const int cA = lane * 4;
    const int cB = 128 + lane * 4;

    // ---- phase 2: per point LN1 + exact GELU + residual feat  (f -> LDS)
    for (int l = wv; l < F_PTS; l += 8) {
        const int g = base + l;
        if (g < 0 || g >= N) continue;             // uniform per wave
        const float4 u0 = *(const float4*)&lin[l][cA];
        const float4 u1 = *(const float4*)&lin[l][cB];
        float v[8] = {u0.x, u0.y, u0.z, u0.w, u1.x, u1.y, u1.z, u1.w};
        float sum = 0.f, sq = 0.f;
#pragma unroll
        for (int j = 0; j < 8; ++j) { sum += v[j]; sq += v[j] * v[j]; }
        sum = wave_sum32(sum);
        sq  = wave_sum32(sq);
        const float mu  = sum * (1.f / C_DIM);
        const float inv = 1.f / sqrtf(sq * (1.f / C_DIM) - mu * mu + 1e-5f);
        const float4 f0 = *(const float4*)(feat + (long long)g * C_DIM + cA);
        const float4 f1 = *(const float4*)(feat + (long long)g * C_DIM + cB);
        const float fe[8] = {f0.x, f0.y, f0.z, f0.w, f1.x, f1.y, f1.z, f1.w};
#pragma unroll
        for (int j = 0; j < 8; ++j) {
            const int ch   = (j < 4) ? (cA + j) : (cB + j - 4);
            const float t  = (v[j] - mu) * inv * g1[ch] + be1[ch];
            const float ge = 0.5f * t * (1.0f + erff(t * 0.70710678118654752f));
            lin[l][ch] = fe[j] + ge;
        }
    }
    __syncthreads();

    // ---- phase 3: depthwise conv k=3 (scene-masked) + residual + LN2
    for (int l = 1 + wv; l < 1 + OUT_PTS; l += 8) {
        const int g = base + l;                    // >= 0 always (l >= 1)
        if (g >= N) continue;                      // uniform per wave
        const float pm = spm[l];
        const float nm = snm[l];

        const float4 p0 = *(const float4*)&lin[l - 1][cA];
        const float4 p1 = *(const float4*)&lin[l - 1][cB];
        const float4 c0 = *(const float4*)&lin[l][cA];
        const float4 c1 = *(const float4*)&lin[l][cB];
        const float4 n0 = *(const float4*)&lin[l + 1][cA];
        const float4 n1 = *(const float4*)&lin[l + 1][cB];
        const float4 f0 = *(const float4*)(feat + (long long)g * C_DIM + cA);
        const float4 f1 = *(const float4*)(feat + (long long)g * C_DIM + cB);
        const float fp[8] = {p0.x, p0.y, p0.z, p0.w, p1.x, p1.y, p1.z, p1.w};
        const float fc[8] = {c0.x, c0.y, c0.z, c0.w, c1.x, c1.y, c1.z, c1.w};
        const float fn[8] = {n0.x, n0.y, n0.z, n0.w, n1.x, n1.y, n1.z, n1.w};
        const float fe[8] = {f0.x, f0.y, f0.z, f0.w, f1.x, f1.y, f1.z, f1.w};

        float r[8];
        float sum = 0.f, sq = 0.f;
#pragma unroll
        for (int j = 0; j < 8; ++j) {
            const int ch = (j < 4) ? (cA + j) : (cB + j - 4);
            const float cv = fp[j] * pm * w_conv[ch * 3 + 0] +
                             fc[j]      * w_conv[ch * 3 + 1] +
                             fn[j] * nm * w_conv[ch * 3 + 2] + b_conv[ch];
            r[j] = fe[j] + cv;
            sum += r[j]; sq += r[j] * r[j];
        }
        sum = wave_sum32(sum);
        sq  = wave_sum32(sq);
        const float mu  = sum * (1.f / C_DIM);
        const float inv = 1.f / sqrtf(sq * (1.f / C_DIM) - mu * mu + 1e-5f);
        float* orow = out + (long long)g * C_DIM;
#pragma unroll
        for (int j = 0; j < 8; ++j) {
            const int ch = (j < 4) ? (cA + j) : (cB + j - 4);
            orow[ch] = (r[j] - mu) * inv * g2[ch] + be2[ch];
        }
    }
}

// ---------------------------------------------------------------------------
extern "C" void kernel_launch(void* const* d_in, const int* in_sizes, int n_in,
                              void* d_out, int out_size, void* d_ws, size_t ws_size,
                              hipStream_t stream) {
    const float*     feat    = (const float*)d_in[0];
    const float*     coord   = (const float*)d_in[1];
    const long long* lengths = (const long long*)d_in[2];
    const float*     w_xyz   = (const float*)d_in[3];
    const float*     b_xyz   = (const float*)d_in[4];
    const float*     g1      = (const float*)d_in[5];
    const float*     be1     = (const float*)d_in[6];
    const float*     w_conv  = (const float*)d_in[7];
    const float*     b_conv  = (const float*)d_in[8];
    const float*     g2      = (const float*)d_in[9];
    const float*     be2     = (const float*)d_in[10];

    const int N  = in_sizes[0] / C_DIM;
    const int Bn = in_sizes[2];
    float* means = (float*)d_ws;    // B*3 floats

    scene_mean_kernel<<<Bn, TPB, 0, stream>>>(coord, lengths, means, Bn);

    const int nblocks = (N + OUT_PTS - 1) / OUT_PTS;
    cpe_fused_kernel<<<nblocks, TPB, 0, stream>>>(
        feat, coord, lengths, w_xyz, b_xyz, g1, be1,
        w_conv, b_conv, g2, be2, means, (float*)d_out, N, Bn);
}